// Loss_S_Probe_40742059770679
// MI455X (gfx1250) — compile-verified
//
#include <hip/hip_runtime.h>
#include <math.h>

typedef __attribute__((ext_vector_type(2))) float v2f;
typedef __attribute__((ext_vector_type(4))) float v4f;
typedef __attribute__((ext_vector_type(8))) float v8f;

// ---------------- Kernel 1: lens[b] = sum(mask[b,:]) - 2 ----------------
__global__ void k_prep(const int* __restrict__ mask, int L, int* __restrict__ lens) {
    const int b   = blockIdx.x;
    const int tid = threadIdx.x;
    int s = 0;
    for (int i = tid; i < L; i += blockDim.x) s += mask[(size_t)b * L + i];
    for (int off = 16; off; off >>= 1) s += __shfl_down(s, off);
    __shared__ int sm[8];
    const int wave = tid >> 5, lane = tid & 31;
    if (lane == 0) sm[wave] = s;
    __syncthreads();
    if (tid == 0) {
        int tot = 0;
        for (int w = 0; w < 8; ++w) tot += sm[w];
        lens[b] = tot - 2;
    }
}

// ---------------- Kernel 2: masked |T - A^2| partial sums ----------------
// Wave w handles row r = blockIdx.y*8 + w of batch blockIdx.z, columns
// [0, rowlen) only (prefix mask). Column loop is split into a branch-free,
// software-pipelined full-block body (2x unconditional nontemporal b128
// loads, 2 shuffles, 4 fma, 1 wmma per 128 columns) and a single predicated
// tail. The +1 input-column shift is realigned in-register: lane l takes
// lane l+1's .x via shfl_down, and lane 31 takes the NEXT block's lane-0 .x
// via a broadcast — no extra scalar loads in the body.
__global__ __launch_bounds__(256) void k_main(const float* __restrict__ inp,
                                              const float* __restrict__ tgt,
                                              const int* __restrict__ lens,
                                              float* __restrict__ blockpart,
                                              int L) {
    const int b    = blockIdx.z;
    const int lane = threadIdx.x & 31;
    const int wave = threadIdx.x >> 5;
    const int r    = blockIdx.y * 8 + wave;

    const int lensv  = lens[b];
    const int rowlen = lensv < 0 ? 0 : lensv;   // <= L-2

    __shared__ float sred[8];
    float waveSum = 0.0f;

    if (r < rowlen) {                            // wave-uniform at runtime
        const float* trow = tgt + ((size_t)b * L + r) * L;
        const float* irow = inp + ((size_t)b * L + (r + 1)) * L;

        v8f acc = {};
        v2f ones; ones[0] = 1.0f; ones[1] = 1.0f;

        const int nFull   = rowlen >> 7;         // full 128-column blocks
        const int tailLen = rowlen & 127;
        const int cl      = lane << 2;           // lane's quad offset

        v4f t4c, i4c, t4n, i4n;

        // ---- tail-stage loader (k == nFull), also supplies the carry for
        //      the last full block via lane 0's i4.x (c = 128*nFull <= rowlen)
        auto loadTail = [&](v4f& t4o, v4f& i4o) {
            const int c = (nFull << 7) + cl;
            v4f tz = {0.f, 0.f, 0.f, 0.f};
            t4o = tz; i4o = tz;
            if (c <  rowlen) t4o = *(const v4f*)(trow + c);   // aligned, in-row
            if (c <= rowlen) i4o = *(const v4f*)(irow + c);   // aligned, in-row
        };

        if (nFull > 0) {
            // prologue: block 0 (full)
            t4c = __builtin_nontemporal_load((const v4f*)(trow + cl));
            i4c = __builtin_nontemporal_load((const v4f*)(irow + cl));

            for (int k = 0; k < nFull - 1; ++k) {
                const int cn = ((k + 1) << 7) + cl;
                t4n = __builtin_nontemporal_load((const v4f*)(trow + cn));
                i4n = __builtin_nontemporal_load((const v4f*)(irow + cn));

                float nx = __shfl_down(i4c.x, 1);
                const float carry = __shfl(i4n.x, 0);   // raw col 128*(k+1)
                nx = (lane == 31) ? carry : nx;

                const float d0 = t4c.x - i4c.y * i4c.y;
                const float d1 = t4c.y - i4c.z * i4c.z;
                const float d2 = t4c.z - i4c.w * i4c.w;
                const float d3 = t4c.w - nx * nx;
                v2f a; a[0] = fabsf(d0) + fabsf(d1); a[1] = fabsf(d2) + fabsf(d3);
                acc = __builtin_amdgcn_wmma_f32_16x16x4_f32(
                    false, a, false, ones, (short)0, acc, false, false);

                t4c = t4n; i4c = i4n;
            }

            // last full block: its carry comes from the tail stage
            loadTail(t4n, i4n);
            {
                float nx = __shfl_down(i4c.x, 1);
                const float carry = __shfl(i4n.x, 0);   // raw col 128*nFull
                nx = (lane == 31) ? carry : nx;

                const float d0 = t4c.x - i4c.y * i4c.y;
                const float d1 = t4c.y - i4c.z * i4c.z;
                const float d2 = t4c.z - i4c.w * i4c.w;
                const float d3 = t4c.w - nx * nx;
                v2f a; a[0] = fabsf(d0) + fabsf(d1); a[1] = fabsf(d2) + fabsf(d3);
                acc = __builtin_amdgcn_wmma_f32_16x16x4_f32(
                    false, a, false, ones, (short)0, acc, false, false);
            }
            t4c = t4n; i4c = i4n;
        } else {
            loadTail(t4c, i4c);
        }

        if (tailLen) {
            const int c = (nFull << 7) + cl;
            float nx = __shfl_down(i4c.x, 1);    // lane 31's col is masked out
            const float in0 = i4c.y, in1 = i4c.z, in2 = i4c.w, in3 = nx;
            const float v0 = (c + 0 < rowlen) ? fabsf(t4c.x - in0 * in0) : 0.0f;
            const float v1 = (c + 1 < rowlen) ? fabsf(t4c.y - in1 * in1) : 0.0f;
            const float v2 = (c + 2 < rowlen) ? fabsf(t4c.z - in2 * in2) : 0.0f;
            const float v3 = (c + 3 < rowlen) ? fabsf(t4c.w - in3 * in3) : 0.0f;
            v2f a; a[0] = v0 + v1; a[1] = v2 + v3;
            acc = __builtin_amdgcn_wmma_f32_16x16x4_f32(
                false, a, false, ones, (short)0, acc, false, false);
        }

        float s = acc[0] + acc[1] + acc[2] + acc[3] +
                  acc[4] + acc[5] + acc[6] + acc[7];
        if ((lane & 15) != 0) s = 0.0f;          // keep column N=0 only (exact)
        for (int off = 16; off; off >>= 1) s += __shfl_down(s, off);
        waveSum = s;                             // valid in lane 0
    }

    if (lane == 0) sred[wave] = waveSum;
    __syncthreads();
    if (threadIdx.x == 0) {
        float t = 0.0f;
        for (int w = 0; w < 8; ++w) t += sred[w];
        blockpart[(size_t)b * gridDim.y + blockIdx.y] = t;   // no atomics
    }
}

// ---------------- Kernel 3: deterministic finalize ----------------
__global__ void k_fin(const float* __restrict__ blockpart,
                      const int* __restrict__ lens,
                      int B, int nrg, float* __restrict__ out) {
    const int wave = threadIdx.x >> 5, lane = threadIdx.x & 31;
    __shared__ float sw[8];
    float acc = 0.0f;
    for (int b = wave; b < B; b += 8) {
        float s = 0.0f;
        for (int i = lane; i < nrg; i += 32)
            s += blockpart[(size_t)b * nrg + i];
        for (int off = 16; off; off >>= 1) s += __shfl_down(s, off);
        if (lane == 0) {
            const int lv = lens[b];
            const int d  = lv * lv > 1 ? lv * lv : 1;
            acc += (lv > 0) ? s / (float)d : 0.0f;
        }
    }
    if (lane == 0) sw[wave] = acc;
    __syncthreads();
    if (threadIdx.x == 0) {
        float t = 0.0f;
        for (int w = 0; w < 8; ++w) t += sw[w];
        out[0] = t / (float)B;
    }
}

extern "C" void kernel_launch(void* const* d_in, const int* in_sizes, int n_in,
                              void* d_out, int out_size, void* d_ws, size_t ws_size,
                              hipStream_t stream) {
    const float* inp  = (const float*)d_in[0];
    const int*   mask = (const int*)d_in[1];
    const float* tgt  = (const float*)d_in[2];
    float*       out  = (float*)d_out;

    const long long tot = (long long)in_sizes[0];
    const int BL = in_sizes[1];
    const int L  = (int)(tot / BL);   // 1024
    const int B  = BL / L;            // 32

    int*   lens      = (int*)d_ws;
    float* blockpart = (float*)d_ws + 64;   // after lens (B <= 64)

    const int rows = L - 1;                 // 1023
    const int nrg  = (rows + 7) / 8;        // row-groups per batch (128)

    k_prep<<<B, 256, 0, stream>>>(mask, L, lens);
    dim3 g(1, nrg, B);
    k_main<<<g, 256, 0, stream>>>(inp, tgt, lens, blockpart, L);
    k_fin<<<1, 256, 0, stream>>>(blockpart, lens, B, nrg, out);
}